// SWD2_28449863369546
// MI455X (gfx1250) — compile-verified
//
#include <hip/hip_runtime.h>
#include <stdint.h>

// Problem constants from the reference: B=2, H=8, L=2048, D=64.
#define L_SEQ 2048
#define D_DIM 64
#define SORT_T 256   // 8 wave32 waves per block

// ---------------------------------------------------------------------------
// CDNA5 async global->LDS copy (ASYNCcnt-tracked). Confirmed lowering to
// global_load_async_to_lds_b32 + s_wait_asynccnt in round 2.
// ---------------------------------------------------------------------------
#if defined(__has_builtin)
#if __has_builtin(__builtin_amdgcn_global_load_async_to_lds_b32)
#define HAVE_ASYNC_LDS 1
#endif
#endif

typedef __attribute__((address_space(1))) int g1_int;  // global
typedef __attribute__((address_space(3))) int l3_int;  // LDS

__device__ __forceinline__ void async_copy_f32(const float* gsrc, float* lds_dst) {
#ifdef HAVE_ASYNC_LDS
  // Flat->AS1 is a same-width inttoptr; flat LDS addresses carry the byte
  // offset in the low 32 bits (ISA 10.2), so truncation yields the AS3 ptr.
  __builtin_amdgcn_global_load_async_to_lds_b32(
      (g1_int*)(uintptr_t)gsrc,
      (l3_int*)(uint32_t)(uintptr_t)lds_dst,
      /*offset=*/0, /*cpol=*/0);
#else
  *lds_dst = *gsrc;   // fallback: plain load + ds_store
#endif
}

__device__ __forceinline__ void wait_async_copies() {
#if defined(__has_builtin) && __has_builtin(__builtin_amdgcn_s_wait_asynccnt)
  __builtin_amdgcn_s_wait_asynccnt(0);
#elif defined(HAVE_ASYNC_LDS)
  asm volatile("s_wait_asynccnt 0" ::: "memory");
#endif
}

// Monotonic float <-> uint32 key mapping (IEEE total order, NaN-free inputs).
__device__ __forceinline__ uint32_t f32_to_key(float f) {
  uint32_t s = __float_as_uint(f);
  return s ^ ((s & 0x80000000u) ? 0xFFFFFFFFu : 0x80000000u);
}
__device__ __forceinline__ float key_to_f32(uint32_t k) {
  uint32_t s = (k & 0x80000000u) ? (k ^ 0x80000000u) : ~k;
  return __uint_as_float(s);
}

// ---------------------------------------------------------------------------
// One block per (bh, d) column:
//  1) async-gather q/k columns into LDS, pack (sortkey<<32 | seq_idx)
//  2) bitonic-argsort both arrays in LDS (stable via idx tiebreak)
//  3) fused scatter: if !mask[bh,qi,ki], atomically add exp(-(qs-ks)^2)/D.
// Untouched cells stay 0 (== reference: masked->0, unmasked-untouched->0/D).
// 1/D = 1/64 is a power of two, so per-term scaling is bit-compatible with
// dividing the sum.
// ---------------------------------------------------------------------------
__global__ __launch_bounds__(SORT_T) void sort_scatter_kernel(
    const float* __restrict__ q, const float* __restrict__ k,
    const int* __restrict__ mask, float* __restrict__ out) {
  __shared__ unsigned long long sq[L_SEQ];   // 16 KB
  __shared__ unsigned long long sk[L_SEQ];   // 16 KB
  __shared__ float stage[L_SEQ];             //  8 KB staging for async loads

  const int col = blockIdx.x;
  const int bh  = col >> 6;           // col / D_DIM
  const int d   = col & (D_DIM - 1);  // col % D_DIM
  const int tid = threadIdx.x;

  const float* qcol = q + (size_t)bh * L_SEQ * D_DIM + d;
  const float* kcol = k + (size_t)bh * L_SEQ * D_DIM + d;

  // ---- q column: async gather into LDS, then pack (key<<32 | seq_idx) ----
  for (int l = tid; l < L_SEQ; l += SORT_T)
    async_copy_f32(qcol + (size_t)l * D_DIM, &stage[l]);
  wait_async_copies();
  __syncthreads();
  for (int l = tid; l < L_SEQ; l += SORT_T)
    sq[l] = ((unsigned long long)f32_to_key(stage[l]) << 32) | (unsigned)l;
  __syncthreads();

  // ---- k column ----
  for (int l = tid; l < L_SEQ; l += SORT_T)
    async_copy_f32(kcol + (size_t)l * D_DIM, &stage[l]);
  wait_async_copies();
  __syncthreads();
  for (int l = tid; l < L_SEQ; l += SORT_T)
    sk[l] = ((unsigned long long)f32_to_key(stage[l]) << 32) | (unsigned)l;
  __syncthreads();

  // ---- bitonic sort both key/idx arrays ---------------------------------
  for (unsigned kw = 2; kw <= L_SEQ; kw <<= 1) {
    for (unsigned j = kw >> 1; j > 0; j >>= 1) {
      for (unsigned t = tid; t < L_SEQ / 2; t += SORT_T) {
        unsigned i   = (t & (j - 1)) | ((t & ~(j - 1)) << 1);
        unsigned ixj = i + j;
        bool up = ((i & kw) == 0);
        unsigned long long a = sq[i], b = sq[ixj];
        if ((a > b) == up) { sq[i] = b; sq[ixj] = a; }
        a = sk[i]; b = sk[ixj];
        if ((a > b) == up) { sk[i] = b; sk[ixj] = a; }
      }
      __syncthreads();
    }
  }

  // ---- fused mask + scale + scatter-add ---------------------------------
  const size_t base = (size_t)bh * L_SEQ * L_SEQ;
  const int*   maskp = mask + base;
  float*       outp  = out + base;
  const float  scale = 1.0f / (float)D_DIM;   // exact (2^-6)

  // Prefetch the random mask gathers first to hide their latency.
  for (int r = tid; r < L_SEQ; r += SORT_T) {
    unsigned qi = (unsigned)(sq[r] & 0xFFFFFFFFull);
    unsigned ki = (unsigned)(sk[r] & 0xFFFFFFFFull);
    __builtin_prefetch(maskp + (size_t)qi * L_SEQ + ki, 0, 0);  // global_prefetch_b8
  }
  for (int r = tid; r < L_SEQ; r += SORT_T) {
    unsigned long long aq = sq[r], ak = sk[r];
    unsigned qi = (unsigned)(aq & 0xFFFFFFFFull);
    unsigned ki = (unsigned)(ak & 0xFFFFFFFFull);
    size_t idx = (size_t)qi * L_SEQ + ki;
    if (maskp[idx] == 0) {
      float qs = key_to_f32((uint32_t)(aq >> 32));
      float ks = key_to_f32((uint32_t)(ak >> 32));
      float dlt = qs - ks;
      float v = __expf(-dlt * dlt) * scale;
      unsafeAtomicAdd(outp + idx, v);   // global_atomic_add_f32
    }
  }
}

extern "C" void kernel_launch(void* const* d_in, const int* in_sizes, int n_in,
                              void* d_out, int out_size, void* d_ws, size_t ws_size,
                              hipStream_t stream) {
  const float* q    = (const float*)d_in[0];
  const float* k    = (const float*)d_in[1];
  const int*   mask = (const int*)d_in[2];   // bool mask delivered as integer
  float* out = (float*)d_out;

  const int BH = in_sizes[0] / (L_SEQ * D_DIM);            // 16
  const long long n = (long long)BH * L_SEQ * L_SEQ;       // 67,108,864

  // 1) zero the accumulator (d_out is poisoned; must re-zero every call)
  (void)hipMemsetAsync(d_out, 0, (size_t)n * sizeof(float), stream);

  // 2) per-(bh,d) sort + fused masked/scaled scatter-add
  sort_scatter_kernel<<<BH * D_DIM, SORT_T, 0, stream>>>(q, k, mask, out);
}